// MAPLoss_73254962201244
// MI455X (gfx1250) — compile-verified
//
#include <hip/hip_runtime.h>
#include <math.h>

// ---------------------------------------------------------------------------
// Smooth-AP loss for MI455X (gfx1250, wave32).
//   query  : [N, D] f32   (N=512, D=256)
//   target : [N]    int32
//   out    : scalar f32 = 1 - meanAP
//
// Phase 1: row-normalize query                    -> qn      [N,D]   (ws)
// Phase 2: Gram matrix qn*qn^T via V_WMMA_F32_16X16X4_F32 -> ranking [N,N] (ws)
//          f32 WMMA chosen deliberately: sigmoid temperature K=0.001 needs
//          similarity resolution finer than f16 ulp (~1e-3 near 1.0).
// Phase 3: fused per-row smooth-AP (LDS-staged off-diag row, v_exp inner loop)
// Phase 4: scalar reduction -> d_out[0]
// ---------------------------------------------------------------------------

typedef __attribute__((ext_vector_type(2))) float v2f;
typedef __attribute__((ext_vector_type(8))) float v8f;

// ---------------- Phase 1: row L2 normalize --------------------------------
__global__ void __launch_bounds__(256)
normalize_kernel(const float* __restrict__ q, float* __restrict__ qn,
                 int N, int D) {
    const int row = blockIdx.x;
    const int tid = threadIdx.x;
    __shared__ float red[256];

    float s = 0.0f;
    for (int c = tid; c < D; c += blockDim.x) {
        float v = q[row * D + c];
        s += v * v;
    }
    red[tid] = s;
    __syncthreads();
    for (int w = blockDim.x >> 1; w > 0; w >>= 1) {
        if (tid < w) red[tid] += red[tid + w];
        __syncthreads();
    }
    const float inv = 1.0f / fmaxf(sqrtf(red[0]), 1e-8f);
    for (int c = tid; c < D; c += blockDim.x)
        qn[row * D + c] = q[row * D + c] * inv;
}

// ---------------- Phase 2: Gram matrix via f32 WMMA ------------------------
// One wave per 16x16 output tile. A-matrix 16x4 f32 layout (ISA 7.12.2):
//   lane L<16 : VGPR0=K0, VGPR1=K1 for row M=L ; lane L>=16 : K2,K3 for M=L-16
// B (4x16, = qn^T slice of a Gram matrix) has the mirror-identical lane loads
// with the column-tile base, so both fragments are one float2 load each.
__global__ void __launch_bounds__(32)
gram_wmma_kernel(const float* __restrict__ qn, float* __restrict__ rank,
                 int N, int D) {
    const int m0   = blockIdx.x * 16;
    const int n0   = blockIdx.y * 16;
    const int lane = threadIdx.x & 31;
    const int r    = lane & 15;
    const int ksel = (lane >> 4) << 1;   // 0 for lanes 0-15, 2 for lanes 16-31

    const float* arow = qn + (size_t)(m0 + r) * D + ksel;
    const float* brow = qn + (size_t)(n0 + r) * D + ksel;

    v8f acc = {};
    for (int k = 0; k < D; k += 4) {
        v2f a = *(const v2f*)(arow + k);   // K = k+ksel, k+ksel+1
        v2f b = *(const v2f*)(brow + k);
        // D = A(16x4,f32) * B(4x16,f32) + C(16x16,f32)
        acc = __builtin_amdgcn_wmma_f32_16x16x4_f32(
            /*neg_a=*/false, a, /*neg_b=*/false, b,
            /*c_mod=*/(short)0, acc, /*reuse_a=*/false, /*reuse_b=*/false);
    }

    // C/D layout: VGPR v -> M = v (+8 for lanes 16-31), N = lane & 15
    const int row_base = m0 + ((lane >> 4) << 3);
    const int col      = n0 + (lane & 15);
#pragma unroll
    for (int v = 0; v < 8; ++v)
        rank[(size_t)(row_base + v) * N + col] = acc[v];
}

// ---------------- Phase 3: fused smooth-AP per query row -------------------
// For query row q with off-diagonal similarities r[0..M-1] and labels g:
//   S_i = sum_j sigma((r_j - r_i)/K), G_i = sum_j g_j sigma(...)
// The diagonal (j==i) term sigma(0)=0.5 is included in the sums and removed
// algebraically: denom = S - 0.5 + 1, acc = G - 0.5*g_i + 1.
__global__ void __launch_bounds__(256)
smoothap_kernel(const float* __restrict__ rank, const int* __restrict__ target,
                float* __restrict__ apv, int N) {
    const int q   = blockIdx.x;
    const int tid = threadIdx.x;
    const int M   = N - 1;

    __shared__ float rs[512];
    __shared__ float gs[512];
    __shared__ float reda[256];
    __shared__ float redn[256];

    const int tq = target[q];
    for (int j = tid; j < M; j += blockDim.x) {
        const int col = (j < q) ? j : j + 1;     // off-diagonal extraction
        rs[j] = rank[(size_t)q * N + col];
        gs[j] = (target[col] == tq) ? 1.0f : 0.0f;
    }
    __syncthreads();

    float ap = 0.0f, npos = 0.0f;
    for (int i = tid; i < M; i += blockDim.x) {
        const float ri = rs[i];
        const float gi = gs[i];
        float S = 0.0f, G = 0.0f;
#pragma unroll 4
        for (int j = 0; j < M; ++j) {
            // sigma((r_j - r_i)*1000) = 1/(1+exp((r_i - r_j)*1000)); inf-safe.
            const float e   = __expf((ri - rs[j]) * 1000.0f);
            const float sig = 1.0f / (1.0f + e);
            S += sig;
            G += gs[j] * sig;
        }
        S += 0.5f;               // (-0.5 diag) + 1.0
        G += 1.0f - 0.5f * gi;   // (-0.5*g_i diag) + 1.0
        ap   += gi * (G / S);
        npos += gi;
    }

    reda[tid] = ap;
    redn[tid] = npos;
    __syncthreads();
    for (int w = blockDim.x >> 1; w > 0; w >>= 1) {
        if (tid < w) { reda[tid] += reda[tid + w]; redn[tid] += redn[tid + w]; }
        __syncthreads();
    }
    if (tid == 0) {
        const float a = reda[0], c = redn[0];
        apv[2 * q]     = (c > 0.0f) ? a / c : 0.0f;   // ap_norm
        apv[2 * q + 1] = (c > 0.0f) ? 1.0f : 0.0f;    // valid flag
    }
}

// ---------------- Phase 4: final scalar ------------------------------------
__global__ void __launch_bounds__(256)
finalize_kernel(const float* __restrict__ apv, float* __restrict__ out, int N) {
    const int tid = threadIdx.x;
    __shared__ float sa[256];
    __shared__ float sc[256];
    float a = 0.0f, c = 0.0f;
    for (int i = tid; i < N; i += blockDim.x) {
        a += apv[2 * i];
        c += apv[2 * i + 1];
    }
    sa[tid] = a; sc[tid] = c;
    __syncthreads();
    for (int w = blockDim.x >> 1; w > 0; w >>= 1) {
        if (tid < w) { sa[tid] += sa[tid + w]; sc[tid] += sc[tid + w]; }
        __syncthreads();
    }
    if (tid == 0)
        out[0] = 1.0f - sa[0] / fmaxf(sc[0], 1.0f);
}

// ---------------------------------------------------------------------------
extern "C" void kernel_launch(void* const* d_in, const int* in_sizes, int n_in,
                              void* d_out, int out_size, void* d_ws, size_t ws_size,
                              hipStream_t stream) {
    const float* query  = (const float*)d_in[0];
    const int*   target = (const int*)d_in[1];

    const int N = in_sizes[1];              // 512
    const int D = in_sizes[0] / N;          // 256

    float* qn   = (float*)d_ws;             // N*D floats
    float* rank = qn + (size_t)N * D;       // N*N floats
    float* apv  = rank + (size_t)N * N;     // 2*N floats
    float* out  = (float*)d_out;

    normalize_kernel<<<N, 256, 0, stream>>>(query, qn, N, D);

    dim3 grid(N / 16, N / 16);
    gram_wmma_kernel<<<grid, 32, 0, stream>>>(qn, rank, N, D);

    smoothap_kernel<<<N, 256, 0, stream>>>(rank, target, apv, N);

    finalize_kernel<<<1, 256, 0, stream>>>(apv, out, N);
}